// CrossAdjacencyMatrix_43843026158044
// MI455X (gfx1250) — compile-verified
//
#include <hip/hip_runtime.h>
#include <hip/hip_bf16.h>

typedef __attribute__((ext_vector_type(2))) float v2f;
typedef __attribute__((ext_vector_type(8))) float v8f;

#define DDIM 128   // feature dim (fixed by reference)

// ---------------------------------------------------------------------------
// Row L2-normalize both relation-weight matrices: out = row / (||row|| + 1e-8)
// One wave (32 lanes) per row; each lane owns a float4 (128 = 32*4).
// ---------------------------------------------------------------------------
__global__ void row_normalize_kernel(const float* __restrict__ A,
                                     const float* __restrict__ B,
                                     float* __restrict__ An,
                                     float* __restrict__ Bn,
                                     int R) {
    int row = blockIdx.x;
    const float* src;
    float* dst;
    if (row < R) { src = A;  dst = An; }
    else         { src = B;  dst = Bn; row -= R; }
    int lane = threadIdx.x;                       // 0..31
    float4 v = reinterpret_cast<const float4*>(src + (size_t)row * DDIM)[lane];
    float ss = v.x * v.x + v.y * v.y + v.z * v.z + v.w * v.w;
    #pragma unroll
    for (int off = 16; off > 0; off >>= 1)
        ss += __shfl_xor(ss, off, 32);
    float inv = 1.0f / (sqrtf(ss) + 1e-8f);
    float4 o = make_float4(v.x * inv, v.y * inv, v.z * inv, v.w * inv);
    reinterpret_cast<float4*>(dst + (size_t)row * DDIM)[lane] = o;
}

// ---------------------------------------------------------------------------
// sim = An @ Bn^T  via V_WMMA_F32_16X16X4_F32 (full-precision f32 matrix core).
// One wave per 16x16 output tile. K=128 -> 32 wmma ops of K=4.
// f32 A-frag (16x4): lane<16 -> row=lane, K={0,1}; lane>=16 -> row=lane-16,
// K={2,3}.  B-frag (4x16) of Bn^T has the identical per-lane pattern on the
// row-major Bn, so both operands are plain coalesced float2 loads.
// ---------------------------------------------------------------------------
__global__ void wmma_gemm_nt_kernel(const float* __restrict__ An,
                                    const float* __restrict__ Bn,
                                    float* __restrict__ sim,
                                    int R) {
    const int tm   = blockIdx.x;
    const int tn   = blockIdx.y;
    const int lane = threadIdx.x;                 // 0..31
    const int r    = lane & 15;
    const int kb   = (lane >> 4) << 1;            // 0 or 2

    const float* ap = An + ((size_t)(tm * 16 + r)) * DDIM + kb;
    const float* bp = Bn + ((size_t)(tn * 16 + r)) * DDIM + kb;

    v8f c = {};
    #pragma unroll 8
    for (int k0 = 0; k0 < DDIM; k0 += 4) {
        float2 af = *reinterpret_cast<const float2*>(ap + k0);
        float2 bf = *reinterpret_cast<const float2*>(bp + k0);
        v2f a; a.x = af.x; a.y = af.y;
        v2f b; b.x = bf.x; b.y = bf.y;
        // 8 args: (neg_a, A, neg_b, B, c_mod, C, reuse_a, reuse_b)
        c = __builtin_amdgcn_wmma_f32_16x16x4_f32(false, a, false, b,
                                                  (short)0, c, false, false);
    }

    // C/D layout: VGPR v, lanes 0-15: M=v, N=lane; lanes 16-31: M=v+8, N=lane-16
    const int n     = lane & 15;
    const int mbase = (lane >> 4) << 3;           // 0 or 8
    float* out = sim + ((size_t)(tm * 16 + mbase)) * R + tn * 16 + n;
    #pragma unroll
    for (int v = 0; v < 8; ++v)
        out[(size_t)v * R] = c[v];
}

// ---------------------------------------------------------------------------
// Max pooling over the sim matrix.
// ---------------------------------------------------------------------------
__global__ void row_max_kernel(const float* __restrict__ sim,
                               float* __restrict__ out, int R) {
    int row  = blockIdx.x;
    int lane = threadIdx.x;
    float m = -2.0f;                              // cosine sims are >= -1
    for (int c = lane; c < R; c += 32)
        m = fmaxf(m, sim[(size_t)row * R + c]);
    #pragma unroll
    for (int off = 16; off > 0; off >>= 1)
        m = fmaxf(m, __shfl_xor(m, off, 32));
    if (lane == 0) out[row] = m;
}

__global__ void col_max_kernel(const float* __restrict__ sim,
                               float* __restrict__ out, int R) {
    int col = blockIdx.x * blockDim.x + threadIdx.x;
    if (col >= R) return;
    float m = -2.0f;
    for (int r = 0; r < R; ++r)
        m = fmaxf(m, sim[(size_t)r * R + col]);
    out[col] = m;
}

// ---------------------------------------------------------------------------
// Edge pipeline. deg starts at 1.0 (identity diagonal contribution folded in).
// ---------------------------------------------------------------------------
__global__ void deg_init_kernel(float* __restrict__ deg, int n) {
    int i = blockIdx.x * blockDim.x + threadIdx.x;
    if (i < n) deg[i] = 1.0f;
}

__global__ void edge_vals_kernel(const float* __restrict__ conf,
                                 const float* __restrict__ imp,
                                 const float* __restrict__ pca,
                                 const int*   __restrict__ rel,
                                 const int*   __restrict__ rows,
                                 const float* __restrict__ rel_w,
                                 float* __restrict__ vals,
                                 float* __restrict__ deg,
                                 int E) {
    int i4 = (blockIdx.x * blockDim.x + threadIdx.x) * 4;
    if (i4 >= E) return;
    if (i4 + 3 < E) {
        float4 cf = *reinterpret_cast<const float4*>(conf + i4);
        float4 im = *reinterpret_cast<const float4*>(imp + i4);
        float4 pc = *reinterpret_cast<const float4*>(pca + i4);
        int4   rl = *reinterpret_cast<const int4*>(rel + i4);
        int4   rw = *reinterpret_cast<const int4*>(rows + i4);
        float v0 = cf.x * im.x * (0.5f * (pc.x + rel_w[rl.x]));
        float v1 = cf.y * im.y * (0.5f * (pc.y + rel_w[rl.y]));
        float v2 = cf.z * im.z * (0.5f * (pc.z + rel_w[rl.z]));
        float v3 = cf.w * im.w * (0.5f * (pc.w + rel_w[rl.w]));
        *reinterpret_cast<float4*>(vals + i4) = make_float4(v0, v1, v2, v3);
        atomicAdd(&deg[rw.x], v0);
        atomicAdd(&deg[rw.y], v1);
        atomicAdd(&deg[rw.z], v2);
        atomicAdd(&deg[rw.w], v3);
    } else {
        for (int j = 0; j < 4 && i4 + j < E; ++j) {
            int i = i4 + j;
            float v = conf[i] * imp[i] * (0.5f * (pca[i] + rel_w[rel[i]]));
            vals[i] = v;
            atomicAdd(&deg[rows[i]], v);
        }
    }
}

__global__ void dinv_kernel(float* __restrict__ deg, int n) {
    int i = blockIdx.x * blockDim.x + threadIdx.x;
    if (i < n) {
        float d = deg[i];
        deg[i] = (d > 0.0f) ? rsqrtf(d) : 0.0f;
    }
}

// Covers E edge entries plus N diagonal entries (diag value = dinv[i]^2).
__global__ void scale_kernel(float* __restrict__ vals,
                             const float* __restrict__ dinv,
                             const int* __restrict__ rows,
                             const int* __restrict__ cols,
                             int E, int n) {
    int i = blockIdx.x * blockDim.x + threadIdx.x;
    if (i < E) {
        vals[i] = vals[i] * dinv[rows[i]] * dinv[cols[i]];
    } else if (i < E + n) {
        float d = dinv[i - E];
        vals[i] = d * d;
    }
}

// ---------------------------------------------------------------------------
extern "C" void kernel_launch(void* const* d_in, const int* in_sizes, int n_in,
                              void* d_out, int out_size, void* d_ws, size_t ws_size,
                              hipStream_t stream) {
    const float* rel_sr_w = (const float*)d_in[0];
    const float* rel_tg_w = (const float*)d_in[1];
    const float* conf_sr  = (const float*)d_in[2];
    const float* imp_sr   = (const float*)d_in[3];
    const float* pca_sr   = (const float*)d_in[4];
    const float* conf_tg  = (const float*)d_in[5];
    const float* imp_tg   = (const float*)d_in[6];
    const float* pca_tg   = (const float*)d_in[7];
    const int*   rel_sr   = (const int*)d_in[8];
    const int*   rel_tg   = (const int*)d_in[9];
    const int*   pos_sr   = (const int*)d_in[10];   // [2, E]
    const int*   pos_tg   = (const int*)d_in[11];

    const int R = in_sizes[0] / DDIM;               // 1024
    const int E = in_sizes[2];                      // 4,000,000
    const int N = out_size / 2 - E;                 // 200,000 entities per side

    float* out_sr = (float*)d_out;                  // [E + N]
    float* out_tg = out_sr + (E + N);               // [E + N]

    // workspace layout
    char* w = (char*)d_ws;
    float* An      = (float*)w;  w += (size_t)R * DDIM * sizeof(float);
    float* Bn      = (float*)w;  w += (size_t)R * DDIM * sizeof(float);
    float* sim     = (float*)w;  w += (size_t)R * R * sizeof(float);
    float* relw_sr = (float*)w;  w += (size_t)R * sizeof(float);
    float* relw_tg = (float*)w;  w += (size_t)R * sizeof(float);
    float* deg_sr  = (float*)w;  w += (size_t)N * sizeof(float);
    float* deg_tg  = (float*)w;  w += (size_t)N * sizeof(float);
    (void)ws_size; (void)n_in;

    // 1) normalize rows of both matrices
    row_normalize_kernel<<<2 * R, 32, 0, stream>>>(rel_sr_w, rel_tg_w, An, Bn, R);

    // 2) sim = An @ Bn^T on the f32 WMMA pipe
    dim3 gemm_grid(R / 16, R / 16);
    wmma_gemm_nt_kernel<<<gemm_grid, 32, 0, stream>>>(An, Bn, sim, R);

    // 3) max pooling
    row_max_kernel<<<R, 32, 0, stream>>>(sim, relw_sr, R);
    col_max_kernel<<<(R + 255) / 256, 256, 0, stream>>>(sim, relw_tg, R);

    const int blk = 256;
    const int gN  = (N + blk - 1) / blk;
    const int gE4 = ((E + 3) / 4 + blk - 1) / blk;
    const int gEN = (E + N + blk - 1) / blk;

    // 4) SR side
    deg_init_kernel<<<gN, blk, 0, stream>>>(deg_sr, N);
    edge_vals_kernel<<<gE4, blk, 0, stream>>>(conf_sr, imp_sr, pca_sr, rel_sr,
                                              pos_sr /*rows*/, relw_sr,
                                              out_sr, deg_sr, E);
    dinv_kernel<<<gN, blk, 0, stream>>>(deg_sr, N);
    scale_kernel<<<gEN, blk, 0, stream>>>(out_sr, deg_sr,
                                          pos_sr /*rows*/, pos_sr + E /*cols*/,
                                          E, N);

    // 5) TG side
    deg_init_kernel<<<gN, blk, 0, stream>>>(deg_tg, N);
    edge_vals_kernel<<<gE4, blk, 0, stream>>>(conf_tg, imp_tg, pca_tg, rel_tg,
                                              pos_tg /*rows*/, relw_tg,
                                              out_tg, deg_tg, E);
    dinv_kernel<<<gN, blk, 0, stream>>>(deg_tg, N);
    scale_kernel<<<gEN, blk, 0, stream>>>(out_tg, deg_tg,
                                          pos_tg /*rows*/, pos_tg + E /*cols*/,
                                          E, N);
}